// MultiHeadAttention_17575006175970
// MI455X (gfx1250) — compile-verified
//
#include <hip/hip_runtime.h>
#include <hip/hip_bf16.h>
#include <cstdint>
#include <cstddef>

#define DM    512
#define HEADS 8
#define DK    64
#define SEQ   2048
#define BATCH 8

#define KC        32     // K-chunk staged in LDS
#define LDS_PITCH 36     // padded k-pitch for transposed [n][k] LDS tiles

typedef __attribute__((ext_vector_type(2))) float v2f;
typedef __attribute__((ext_vector_type(4))) float v4f;
typedef __attribute__((ext_vector_type(8))) float v8f;

// ---- async global->LDS path (gfx1250 ASYNCcnt engine), guarded -------------
#ifdef __has_builtin
#if __has_builtin(__builtin_amdgcn_global_load_async_to_lds_b32) && \
    __has_builtin(__builtin_amdgcn_global_load_async_to_lds_b128)
#define USE_ASYNC_LDS 1
#endif
#endif
#ifndef USE_ASYNC_LDS
#define USE_ASYNC_LDS 0
#endif

typedef int vsi4 __attribute__((vector_size(16)));

// signature (from hipcc diagnostics): (AS1 T*, AS3 T*, imm offset, imm cpol)
#define AS1I(p)  ((__attribute__((address_space(1))) int*)(uintptr_t)(p))
#define AS3I(p)  ((__attribute__((address_space(3))) int*)(unsigned int)(uintptr_t)(p))
#define AS1V4(p) ((__attribute__((address_space(1))) vsi4*)(uintptr_t)(p))
#define AS3V4(p) ((__attribute__((address_space(3))) vsi4*)(unsigned int)(uintptr_t)(p))

__device__ __forceinline__ void wait_async0() {
#if USE_ASYNC_LDS
#if __has_builtin(__builtin_amdgcn_s_wait_asynccnt)
  __builtin_amdgcn_s_wait_asynccnt(0);
#else
  asm volatile("s_wait_asynccnt 0" ::: "memory");
#endif
#endif
}

// D = A(16x4, f32) * B(4x16, f32) + C(16x16, f32), full f32 precision.
__device__ __forceinline__ v8f wmma_f32(v2f a, v2f b, v8f c) {
  return __builtin_amdgcn_wmma_f32_16x16x4_f32(
      /*neg_a=*/false, a, /*neg_b=*/false, b,
      /*c_mod=*/(short)0, c, /*reuse_a=*/false, /*reuse_b=*/false);
}

// ---------------------------------------------------------------------------
// GEMM, K = 512: C[M,512] = A[M,512] @ W[512,512] (+ R)
// Block: 256 threads = 8 waves; wave computes 16(M) x 64(N); block 128 x 64.
// W tile staged transposed in LDS, double buffered, software pipelined.
// Grid: (M/128, 512/64)
// ---------------------------------------------------------------------------
template <bool RESID>
__global__ void gemm_k512(const float* __restrict__ A, const float* __restrict__ W,
                          const float* __restrict__ R, float* __restrict__ C) {
  __shared__ float ldsB[2][64 * LDS_PITCH];

  const int t     = threadIdx.x;
  const int lane  = t & 31;
  const int wave  = t >> 5;
  const int m16   = lane & 15;
  const int kh    = (lane >> 4) << 1;   // 0 or 2
  const int tileM = blockIdx.x * 128 + wave * 16;
  const int tileN = blockIdx.y * 64;

  // loader mapping: thread covers k=lk (one row of chunk), n = ln..ln+7
  const int lk = t >> 3;          // 0..31
  const int ln = (t & 7) << 3;    // 0,8,...,56

  v8f acc[4] = {};
  const float* arow = A + (size_t)(tileM + m16) * DM;

  {  // prologue: stage chunk 0 (transposed)
    const float* src = W + (size_t)lk * DM + tileN + ln;
    float* dst = &ldsB[0][0];
#if USE_ASYNC_LDS
#pragma unroll
    for (int i = 0; i < 8; ++i)
      __builtin_amdgcn_global_load_async_to_lds_b32(
          AS1I(src + i), AS3I(dst + (ln + i) * LDS_PITCH + lk), 0, 0);
#else
    v4f w0 = *(const v4f*)(src);
    v4f w1 = *(const v4f*)(src + 4);
#pragma unroll
    for (int i = 0; i < 4; ++i) dst[(ln + i) * LDS_PITCH + lk] = w0[i];
#pragma unroll
    for (int i = 0; i < 4; ++i) dst[(ln + 4 + i) * LDS_PITCH + lk] = w1[i];
#endif
  }
  wait_async0();
  __syncthreads();

  const int NC = DM / KC;  // 16 chunks
  for (int c = 0; c < NC; ++c) {
    const int k0 = c * KC;

#if USE_ASYNC_LDS
    if (c + 1 < NC) {  // fire-and-forget stage of next chunk
      const float* src = W + (size_t)(k0 + KC + lk) * DM + tileN + ln;
      float* dst = &ldsB[(c + 1) & 1][0];
#pragma unroll
      for (int i = 0; i < 8; ++i)
        __builtin_amdgcn_global_load_async_to_lds_b32(
            AS1I(src + i), AS3I(dst + (ln + i) * LDS_PITCH + lk), 0, 0);
    }
#else
    v4f w0 = {}, w1 = {};
    if (c + 1 < NC) {  // issue next chunk's global loads into registers
      const float* src = W + (size_t)(k0 + KC + lk) * DM + tileN + ln;
      w0 = *(const v4f*)(src);
      w1 = *(const v4f*)(src + 4);
    }
#endif

    // preload all A fragments for this chunk (clause of b64 loads, same base)
    v2f afr[KC / 4];
#pragma unroll
    for (int ks = 0; ks < KC / 4; ++ks)
      afr[ks] = *(const v2f*)(arow + k0 + ks * 4 + kh);

    const float* bbuf = &ldsB[c & 1][0];
#pragma unroll
    for (int ks = 0; ks < KC / 4; ++ks) {
      const int kk = ks * 4 + kh;
#pragma unroll
      for (int j = 0; j < 4; ++j) {
        v2f bf = *(const v2f*)(bbuf + (j * 16 + m16) * LDS_PITCH + kk);
        acc[j] = wmma_f32(afr[ks], bf, acc[j]);
      }
    }

#if !USE_ASYNC_LDS
    if (c + 1 < NC) {  // deposit staged registers into the other buffer
      float* dst = &ldsB[(c + 1) & 1][0];
#pragma unroll
      for (int i = 0; i < 4; ++i) dst[(ln + i) * LDS_PITCH + lk] = w0[i];
#pragma unroll
      for (int i = 0; i < 4; ++i) dst[(ln + 4 + i) * LDS_PITCH + lk] = w1[i];
    }
#endif
    wait_async0();
    __syncthreads();
  }

  const int rbase = tileM + ((lane >> 4) << 3);
#pragma unroll
  for (int j = 0; j < 4; ++j) {
    const int col = tileN + j * 16 + m16;
#pragma unroll
    for (int r = 0; r < 8; ++r) {
      const size_t idx = (size_t)(rbase + r) * DM + col;
      float v = acc[j][r];
      if (RESID) v += R[idx];
      C[idx] = v;
    }
  }
}

// ---------------------------------------------------------------------------
// scores[bh, m, n] = (1/8) * sum_k Q[b,m,h*64+k] * K[b,n,h*64+k]  (+ mask)
// K tile rows contiguous along k -> async b128 stages into transposed layout.
// K = 64 -> 2 chunks. Grid: (SEQ/128, SEQ/64, B*H)
// ---------------------------------------------------------------------------
__global__ void scores_kernel(const float* __restrict__ Q, const float* __restrict__ Kmat,
                              const uint8_t* __restrict__ mask, float* __restrict__ attn) {
  __shared__ float ldsB[2][64 * LDS_PITCH];

  const int t     = threadIdx.x;
  const int lane  = t & 31;
  const int wave  = t >> 5;
  const int m16   = lane & 15;
  const int kh    = (lane >> 4) << 1;
  const int bh    = blockIdx.z;
  const int b     = bh >> 3;
  const int h     = bh & 7;
  const int tileM = blockIdx.x * 128 + wave * 16;
  const int tileN = blockIdx.y * 64;

  // loader mapping: thread covers n = lnn, k = lkc..lkc+7
  const int lnn = t >> 2;         // 0..63
  const int lkc = (t & 3) << 3;   // 0,8,16,24

  const float* ksrc = Kmat + ((size_t)b * SEQ + tileN + lnn) * DM + h * DK + lkc;
  const float* qrow = Q + ((size_t)b * SEQ + tileM + m16) * DM + h * DK;

  v8f acc[4] = {};

  {  // prologue: stage chunk 0 (k = 0..31)
    float* dst = &ldsB[0][lnn * LDS_PITCH + lkc];
#if USE_ASYNC_LDS
    __builtin_amdgcn_global_load_async_to_lds_b128(AS1V4(ksrc),     AS3V4(dst),     0, 0);
    __builtin_amdgcn_global_load_async_to_lds_b128(AS1V4(ksrc + 4), AS3V4(dst + 4), 0, 0);
#else
    v4f k0v = *(const v4f*)(ksrc);
    v4f k1v = *(const v4f*)(ksrc + 4);
    *(v4f*)(dst)     = k0v;
    *(v4f*)(dst + 4) = k1v;
#endif
  }
  wait_async0();
  __syncthreads();

  const int NC = DK / KC;  // 2 chunks
#pragma unroll
  for (int c = 0; c < NC; ++c) {
    const int k0 = c * KC;

#if USE_ASYNC_LDS
    if (c + 1 < NC) {
      float* dst = &ldsB[(c + 1) & 1][lnn * LDS_PITCH + lkc];
      __builtin_amdgcn_global_load_async_to_lds_b128(AS1V4(ksrc + KC),     AS3V4(dst),     0, 0);
      __builtin_amdgcn_global_load_async_to_lds_b128(AS1V4(ksrc + KC + 4), AS3V4(dst + 4), 0, 0);
    }
#else
    v4f k0v = {}, k1v = {};
    if (c + 1 < NC) {
      k0v = *(const v4f*)(ksrc + KC);
      k1v = *(const v4f*)(ksrc + KC + 4);
    }
#endif

    v2f afr[KC / 4];
#pragma unroll
    for (int ks = 0; ks < KC / 4; ++ks)
      afr[ks] = *(const v2f*)(qrow + k0 + ks * 4 + kh);

    const float* bbuf = &ldsB[c & 1][0];
#pragma unroll
    for (int ks = 0; ks < KC / 4; ++ks) {
      const int kk = ks * 4 + kh;
#pragma unroll
      for (int j = 0; j < 4; ++j) {
        v2f bf = *(const v2f*)(bbuf + (j * 16 + m16) * LDS_PITCH + kk);
        acc[j] = wmma_f32(afr[ks], bf, acc[j]);
      }
    }

#if !USE_ASYNC_LDS
    if (c + 1 < NC) {
      float* dst = &ldsB[(c + 1) & 1][lnn * LDS_PITCH + lkc];
      *(v4f*)(dst)     = k0v;
      *(v4f*)(dst + 4) = k1v;
    }
#endif
    wait_async0();
    __syncthreads();
  }

  const float scale = 0.125f;  // 1/sqrt(64)
  const int rbase = tileM + ((lane >> 4) << 3);
  float* abh = attn + (size_t)bh * SEQ * SEQ;
  const uint8_t* mb = mask + (size_t)b * SEQ * SEQ;
#pragma unroll
  for (int j = 0; j < 4; ++j) {
    const int col = tileN + j * 16 + m16;
#pragma unroll
    for (int r = 0; r < 8; ++r) {
      const int row = rbase + r;
      float v = acc[j][r] * scale;
      if (mb[(size_t)row * SEQ + col]) v = -1.0e9f;
      abh[(size_t)row * SEQ + col] = v;
    }
  }
}

// ---------------------------------------------------------------------------
// In-place row softmax over 2048 columns. One 256-thread block per row.
// ---------------------------------------------------------------------------
__global__ void softmax_kernel(float* __restrict__ attn) {
  __shared__ float red[256];
  const int t = threadIdx.x;
  float* row = attn + (size_t)blockIdx.x * SEQ;

  float x[8];
  float mx = -3.0e38f;
#pragma unroll
  for (int i = 0; i < 8; ++i) { x[i] = row[t + i * 256]; mx = fmaxf(mx, x[i]); }
  red[t] = mx; __syncthreads();
  for (int s = 128; s > 0; s >>= 1) {
    if (t < s) red[t] = fmaxf(red[t], red[t + s]);
    __syncthreads();
  }
  mx = red[0]; __syncthreads();

  float sum = 0.0f;
#pragma unroll
  for (int i = 0; i < 8; ++i) { x[i] = __expf(x[i] - mx); sum += x[i]; }
  red[t] = sum; __syncthreads();
  for (int s = 128; s > 0; s >>= 1) {
    if (t < s) red[t] += red[t + s];
    __syncthreads();
  }
  const float inv = 1.0f / red[0];
#pragma unroll
  for (int i = 0; i < 8; ++i) row[t + i * 256] = x[i] * inv;
}

// ---------------------------------------------------------------------------
// ctx[b, m, h*64+n] = sum_k attn[bh, m, k] * V[b, k, h*64+n]
// V tile staged transposed in LDS. K = 2048 -> 64 chunks, 2048 WMMA per wave.
// Grid: (SEQ/128, 1, B*H)
// ---------------------------------------------------------------------------
__global__ void context_kernel(const float* __restrict__ attn, const float* __restrict__ V,
                               float* __restrict__ ctx) {
  __shared__ float ldsB[2][64 * LDS_PITCH];

  const int t     = threadIdx.x;
  const int lane  = t & 31;
  const int wave  = t >> 5;
  const int m16   = lane & 15;
  const int kh    = (lane >> 4) << 1;
  const int bh    = blockIdx.z;
  const int b     = bh >> 3;
  const int h     = bh & 7;
  const int tileM = blockIdx.x * 128 + wave * 16;

  const int lk = t >> 3;          // chunk row 0..31
  const int ln = (t & 7) << 3;    // n 0..56 step 8

  const float* arow  = attn + ((size_t)bh * SEQ + tileM + m16) * SEQ;
  const float* vbase = V + (size_t)b * SEQ * DM + h * DK;

  v8f acc[4] = {};

  {  // prologue: stage chunk 0
    const float* src = vbase + (size_t)lk * DM + ln;
    float* dst = &ldsB[0][0];
#if USE_ASYNC_LDS
#pragma unroll
    for (int i = 0; i < 8; ++i)
      __builtin_amdgcn_global_load_async_to_lds_b32(
          AS1I(src + i), AS3I(dst + (ln + i) * LDS_PITCH + lk), 0, 0);
#else
    v4f w0 = *(const v4f*)(src);
    v4f w1 = *(const v4f*)(src + 4);
#pragma unroll
    for (int i = 0; i < 4; ++i) dst[(ln + i) * LDS_PITCH + lk] = w0[i];
#pragma unroll
    for (int i = 0; i < 4; ++i) dst[(ln + 4 + i) * LDS_PITCH + lk] = w1[i];
#endif
  }
  wait_async0();
  __syncthreads();

  const int NC = SEQ / KC;  // 64 chunks
  for (int c = 0; c < NC; ++c) {
    const int k0 = c * KC;

#if USE_ASYNC_LDS
    if (c + 1 < NC) {
      const float* src = vbase + (size_t)(k0 + KC + lk) * DM + ln;
      float* dst = &ldsB[(c + 1) & 1][0];
#pragma unroll
      for (int i = 0; i < 8; ++i)
        __builtin_amdgcn_global_load_async_to_lds_b32(
            AS1I(src + i), AS3I(dst + (ln + i) * LDS_PITCH + lk), 0, 0);
    }
#else
    v4f w0 = {}, w1 = {};
    if (c + 1 < NC) {
      const float* src = vbase + (size_t)(k0 + KC + lk) * DM + ln;
      w0 = *(const v4f*)(src);
      w1 = *(const v4f*)(src + 4);
    }
#endif

    v2f afr[KC / 4];
#pragma unroll
    for (int ks = 0; ks < KC / 4; ++ks)
      afr[ks] = *(const v2f*)(arow + k0 + ks * 4 + kh);

    const float* bbuf = &ldsB[c & 1][0];
#pragma unroll
    for (int ks = 0; ks < KC / 4; ++ks) {
      const int kk = ks * 4 + kh;
#pragma unroll
      for (int j = 0; j < 4; ++j) {
        v2f bf = *(const v2f*)(bbuf + (j * 16 + m16) * LDS_PITCH + kk);
        acc[j] = wmma_f32(afr[ks], bf, acc[j]);
      }
    }

#if !USE_ASYNC_LDS
    if (c + 1 < NC) {
      float* dst = &ldsB[(c + 1) & 1][0];
#pragma unroll
      for (int i = 0; i < 4; ++i) dst[(ln + i) * LDS_PITCH + lk] = w0[i];
#pragma unroll
      for (int i = 0; i < 4; ++i) dst[(ln + 4 + i) * LDS_PITCH + lk] = w1[i];
    }
#endif
    wait_async0();
    __syncthreads();
  }

  const int rbase = tileM + ((lane >> 4) << 3);
#pragma unroll
  for (int j = 0; j < 4; ++j) {
    const int col = h * DK + j * 16 + m16;
#pragma unroll
    for (int r = 0; r < 8; ++r)
      ctx[((size_t)b * SEQ + rbase + r) * DM + col] = acc[j][r];
  }
}

// ---------------------------------------------------------------------------
// Row LayerNorm over 512 columns (weight=1, bias=0, eps=1e-5).
// ---------------------------------------------------------------------------
__global__ void layernorm_kernel(const float* __restrict__ y, float* __restrict__ out) {
  __shared__ float red[256];
  const int t = threadIdx.x;
  const float* row = y + (size_t)blockIdx.x * DM;
  const float a = row[t];
  const float c = row[t + 256];

  red[t] = a + c; __syncthreads();
  for (int s = 128; s > 0; s >>= 1) {
    if (t < s) red[t] += red[t + s];
    __syncthreads();
  }
  const float mu = red[0] * (1.0f / 512.0f); __syncthreads();

  const float da = a - mu, dc = c - mu;
  red[t] = da * da + dc * dc; __syncthreads();
  for (int s = 128; s > 0; s >>= 1) {
    if (t < s) red[t] += red[t + s];
    __syncthreads();
  }
  const float rstd = rsqrtf(red[0] * (1.0f / 512.0f) + 1e-5f);

  float* orow = out + (size_t)blockIdx.x * DM;
  orow[t]       = da * rstd;
  orow[t + 256] = dc * rstd;
}

// ---------------------------------------------------------------------------
extern "C" void kernel_launch(void* const* d_in, const int* in_sizes, int n_in,
                              void* d_out, int out_size, void* d_ws, size_t ws_size,
                              hipStream_t stream) {
  const float*   inQ  = (const float*)d_in[0];
  const float*   inK  = (const float*)d_in[1];
  const float*   inV  = (const float*)d_in[2];
  const uint8_t* mask = (const uint8_t*)d_in[3];
  const float*   WQ   = (const float*)d_in[4];
  const float*   WK   = (const float*)d_in[5];
  const float*   WV   = (const float*)d_in[6];
  const float*   WO   = (const float*)d_in[7];

  float* out  = (float*)d_out;                        // [B,S,512]
  float* attn = out + (size_t)BATCH * SEQ * DM;       // [B,H,S,S]

  const size_t qkv = (size_t)BATCH * SEQ * DM;
  float* Q   = (float*)d_ws;
  float* Kp  = Q + qkv;
  float* V   = Kp + qkv;
  float* ctx = Q;    // Q dead after scores_kernel
  float* y   = Kp;   // K dead after scores_kernel

  const dim3 blk(256);
  const dim3 gProj((BATCH * SEQ) / 128, DM / 64);     // (128, 8)

  gemm_k512<false><<<gProj, blk, 0, stream>>>(inQ, WQ, nullptr, Q);
  gemm_k512<false><<<gProj, blk, 0, stream>>>(inK, WK, nullptr, Kp);
  gemm_k512<false><<<gProj, blk, 0, stream>>>(inV, WV, nullptr, V);

  const dim3 gScores(SEQ / 128, SEQ / 64, BATCH * HEADS);  // (16, 32, 64)
  scores_kernel<<<gScores, blk, 0, stream>>>(Q, Kp, mask, attn);

  softmax_kernel<<<dim3(BATCH * HEADS * SEQ), blk, 0, stream>>>(attn);

  const dim3 gCtx(SEQ / 128, 1, BATCH * HEADS);
  context_kernel<<<gCtx, blk, 0, stream>>>(attn, V, ctx);

  gemm_k512<true><<<gProj, blk, 0, stream>>>(ctx, WO, inQ, y);

  layernorm_kernel<<<dim3(BATCH * SEQ), blk, 0, stream>>>(y, out);
}